// Net_46256797778021
// MI455X (gfx1250) — compile-verified
//
#include <hip/hip_runtime.h>
#include <hip/hip_bf16.h>
#include <math.h>

// ---------- types ----------
typedef float v2f __attribute__((ext_vector_type(2)));
typedef float v8f __attribute__((ext_vector_type(8)));

// ---------- small utility kernels ----------
__global__ void k_fill(float* __restrict__ p, float v, int n) {
    int i = blockIdx.x * blockDim.x + threadIdx.x;
    if (i < n) p[i] = v;
}

__global__ void k_copy(float* __restrict__ dstp, const float* __restrict__ srcp, int n) {
    int i = blockIdx.x * blockDim.x + threadIdx.x;
    if (i < n) dstp[i] = srcp[i];
}

// deg[d] += 1 for every edge destination (self-loop folded in via k_fill(deg,1))
__global__ void k_count_deg(const int* __restrict__ dst, float* __restrict__ deg, int e) {
    int i = blockIdx.x * blockDim.x + threadIdx.x;
    if (i < e) atomicAdd(&deg[dst[i]], 1.0f);
}

__global__ void k_dinv(const float* __restrict__ deg, float* __restrict__ dinv, int n) {
    int i = blockIdx.x * blockDim.x + threadIdx.x;
    if (i < n) {
        float d = deg[i];
        dinv[i] = (d > 0.0f) ? rsqrtf(d) : 0.0f;
    }
}

// ---------- GCN layer kernels ----------
// Per node: optional ReLU on input, h = x @ W.T, and initialize the output
// accumulator with bias + self-loop contribution dinv[i]^2 * h[i].
template <int CIN, int COUT, bool RELU_IN>
__global__ void k_gcn_node(const float* __restrict__ xin, const float* __restrict__ W,
                           const float* __restrict__ b, const float* __restrict__ dinv,
                           float* __restrict__ h, float* __restrict__ out, int n) {
    int i = blockIdx.x * blockDim.x + threadIdx.x;
    if (i >= n) return;
    float xr[CIN];
#pragma unroll
    for (int c = 0; c < CIN; ++c) {
        float v = xin[i * CIN + c];
        xr[c] = RELU_IN ? fmaxf(v, 0.0f) : v;
    }
    float di = dinv[i];
    float dii = di * di;
#pragma unroll
    for (int o = 0; o < COUT; ++o) {
        float hv = 0.0f;
#pragma unroll
        for (int c = 0; c < CIN; ++c) hv = fmaf(xr[c], W[o * CIN + c], hv);
        h[i * COUT + o] = hv;
        out[i * COUT + o] = b[o] + dii * hv;
    }
}

// Per edge: out[d] += dinv[s]*dinv[d] * h[s]
template <int COUT>
__global__ void k_gcn_edge(const int* __restrict__ src, const int* __restrict__ dst,
                           const float* __restrict__ dinv, const float* __restrict__ h,
                           float* __restrict__ out, int e) {
    int i = blockIdx.x * blockDim.x + threadIdx.x;
    if (i >= e) return;
    int s = src[i], d = dst[i];
    float norm = dinv[s] * dinv[d];
#pragma unroll
    for (int o = 0; o < COUT; ++o)
        atomicAdd(&out[d * COUT + o], norm * h[s * COUT + o]);
}

// ---------- big GEMV: y += Wa @ x via V_WMMA_F32_16X16X4_F32 ----------
// Grid: (H/16) blocks of 256 threads (8 waves). Each wave handles the block's
// 16 rows over a kc-column slice using fp32 16x16x4 WMMA:
//   A(16x4): lane L<16 holds Wa[row0+L, k..k+1]; lane L>=16 holds k+2..k+3
//   B(4x16): x[k..k+3] replicated to every column (mirrored lane split)
// Every D column is identical; lanes 0 / 16 hold rows row0+v / row0+8+v.
// kc is precomputed on the host (multiple of 16) -> no runtime division.
__global__ __launch_bounds__(256) void k_gemv_wmma(const float* __restrict__ Wa,
                                                   const float* __restrict__ x,
                                                   float* __restrict__ y,
                                                   int K, int kc) {
    const int lane = threadIdx.x & 31;
    const int wave = threadIdx.x >> 5;
    const int row0 = blockIdx.x * 16;
    const int k0 = wave * kc;
    const int hi = (lane >= 16) ? 2 : 0;
    const int r = row0 + (lane & 15);

    const float* wp = Wa + (size_t)r * (size_t)K + hi;
    const float* xp = x + hi;

    v8f acc = {0.f, 0.f, 0.f, 0.f, 0.f, 0.f, 0.f, 0.f};
    // 16 columns (4 WMMA k-steps) per outer iteration; kc is a multiple of 16.
    for (int k = k0; k < k0 + kc; k += 16) {
        // prefetch the Wa stream two 128B-line blocks ahead
        __builtin_prefetch(wp + k + 64, 0, 1);
#pragma unroll
        for (int kk = 0; kk < 16; kk += 4) {
            float2 w2 = *(const float2*)(wp + k + kk);
            float2 x2 = *(const float2*)(xp + k + kk);
            v2f a; a[0] = w2.x; a[1] = w2.y;
            v2f b; b[0] = x2.x; b[1] = x2.y;
            acc = __builtin_amdgcn_wmma_f32_16x16x4_f32(
                /*neg_a=*/false, a, /*neg_b=*/false, b,
                /*c_mod=*/(short)0, acc, /*reuse_a=*/false, /*reuse_b=*/false);
        }
    }
    if ((lane & 15) == 0) {
        int base = row0 + ((lane >= 16) ? 8 : 0);
#pragma unroll
        for (int v = 0; v < 8; ++v) atomicAdd(&y[base + v], acc[v]);
    }
}

// ---------- final: out[o] = tanh( sum_k Wb[o,k]*tanh(y[k]) + bb[o] ) ----------
__global__ __launch_bounds__(256) void k_final(const float* __restrict__ Wb,
                                               const float* __restrict__ bb,
                                               const float* __restrict__ y,
                                               float* __restrict__ out, int H) {
    __shared__ float red[256];
    int o = blockIdx.x;
    float s = 0.0f;
    for (int k = threadIdx.x; k < H; k += 256)
        s = fmaf(Wb[(size_t)o * H + k], tanhf(y[k]), s);
    red[threadIdx.x] = s;
    __syncthreads();
    for (int w = 128; w > 0; w >>= 1) {
        if (threadIdx.x < w) red[threadIdx.x] += red[threadIdx.x + w];
        __syncthreads();
    }
    if (threadIdx.x == 0) out[o] = tanhf(red[0] + bb[o]);
}

// ---------- launcher ----------
extern "C" void kernel_launch(void* const* d_in, const int* in_sizes, int n_in,
                              void* d_out, int out_size, void* d_ws, size_t ws_size,
                              hipStream_t stream) {
    const int N = in_sizes[0] / 2;   // data [N,2]
    const int E = in_sizes[1] / 2;   // edge [2,E]
    const int H = in_sizes[9];       // ba size = 2N
    const int ONUM = in_sizes[11];   // bb size = OUT

    const float* data = (const float*)d_in[0];
    const int*   edge = (const int*)d_in[1];
    const int*   src  = edge;
    const int*   dst  = edge + E;
    const float* W1 = (const float*)d_in[2];  const float* b1 = (const float*)d_in[3];
    const float* W2 = (const float*)d_in[4];  const float* b2 = (const float*)d_in[5];
    const float* W3 = (const float*)d_in[6];  const float* b3 = (const float*)d_in[7];
    const float* Wa = (const float*)d_in[8];  const float* ba = (const float*)d_in[9];
    const float* Wb = (const float*)d_in[10]; const float* bb = (const float*)d_in[11];

    // workspace layout (floats): 18N total
    float* ws   = (float*)d_ws;
    float* deg  = ws;              // N
    float* dinv = deg  + N;        // N
    float* h1   = dinv + N;        // 4N
    float* o1   = h1   + 4 * N;    // 4N
    float* h2   = o1   + 4 * N;    // 2N
    float* o2   = h2   + 2 * N;    // 2N
    float* h3   = o2   + 2 * N;    // N
    float* o3   = h3   + N;        // N
    float* y    = o3   + N;        // H = 2N

    const int T = 256;
    const int NB = (N + T - 1) / T;
    const int EB = (E + T - 1) / T;

    // degrees (self-loop contributes the initial 1), then deg^-1/2
    k_fill<<<NB, T, 0, stream>>>(deg, 1.0f, N);
    k_count_deg<<<EB, T, 0, stream>>>(dst, deg, E);
    k_dinv<<<NB, T, 0, stream>>>(deg, dinv, N);

    // GCN layer 1: 2 -> 4 (no input relu), output kept pre-relu
    k_gcn_node<2, 4, false><<<NB, T, 0, stream>>>(data, W1, b1, dinv, h1, o1, N);
    k_gcn_edge<4><<<EB, T, 0, stream>>>(src, dst, dinv, h1, o1, E);

    // GCN layer 2: 4 -> 2 (relu applied on read)
    k_gcn_node<4, 2, true><<<NB, T, 0, stream>>>(o1, W2, b2, dinv, h2, o2, N);
    k_gcn_edge<2><<<EB, T, 0, stream>>>(src, dst, dinv, h2, o2, E);

    // GCN layer 3: 2 -> 1 (relu applied on read), no output relu
    k_gcn_node<2, 1, true><<<NB, T, 0, stream>>>(o2, W3, b3, dinv, h3, o3, N);
    k_gcn_edge<1><<<EB, T, 0, stream>>>(src, dst, dinv, h3, o3, E);

    // y = ba; y += Wa @ o3  (WMMA GEMV, K split over 8 waves/block)
    k_copy<<<(H + T - 1) / T, T, 0, stream>>>(y, ba, H);
    const int kc = N / 8;            // per-wave K slice; N=8192 -> 1024 (mult of 16)
    k_gemv_wmma<<<H / 16, T, 0, stream>>>(Wa, o3, y, N, kc);

    // out = tanh(Wb @ tanh(y) + bb)
    k_final<<<ONUM, T, 0, stream>>>(Wb, bb, y, (float*)d_out, H);
}